// SpatialAttention_19602230739048
// MI455X (gfx1250) — compile-verified
//
#include <hip/hip_runtime.h>
#include <math.h>

typedef float v2f __attribute__((ext_vector_type(2)));
typedef float v8f __attribute__((ext_vector_type(8)));

#define Bn 16
#define Cn 64
#define Hn 256
#define Wn 256
#define HW (Hn * Wn)      // 65536
#define CHW (Cn * HW)     // 4194304

// ---------------- Kernel A: channel mean & max (bandwidth-bound) -------------
// 262144 threads, each owns 4 consecutive pixels (float4), loops 64 channels.
__global__ __launch_bounds__(256) void sa_reduce_kernel(
    const float* __restrict__ x, float* __restrict__ avgp, float* __restrict__ maxp) {
  int gid = blockIdx.x * 256 + threadIdx.x;   // 0 .. 262143
  int b   = gid >> 14;                        // / (HW/4)
  int p4  = gid & 16383;
  const float4* base = (const float4*)(x + (size_t)b * CHW) + p4;
  float4 s = base[0];
  float4 m = s;
  #pragma unroll 8
  for (int c = 1; c < Cn; ++c) {
    float4 v = base[(size_t)c * (HW / 4)];
    s.x += v.x; s.y += v.y; s.z += v.z; s.w += v.w;
    m.x = fmaxf(m.x, v.x); m.y = fmaxf(m.y, v.y);
    m.z = fmaxf(m.z, v.z); m.w = fmaxf(m.w, v.w);
  }
  const float inv = 1.0f / (float)Cn;
  s.x *= inv; s.y *= inv; s.z *= inv; s.w *= inv;
  ((float4*)(avgp + (size_t)b * HW))[p4] = s;
  ((float4*)(maxp + (size_t)b * HW))[p4] = m;
}

// ---------------- Kernel B: 7x7 conv + bias + sigmoid via WMMA ---------------
// Block = 256 threads = 8 waves; block covers 128 consecutive x of one row y.
// Each wave computes 16 output pixels with 28 x V_WMMA_F32_16X16X4_F32:
//   A (16x4)  = conv weights broadcast across M  -> every D row identical
//   B (4x16)  = im2col patch, N = pixel
// K is tiled over taps laid out as [c][ky][kx padded to 8] (pad weight = 0).
// f32 A layout (ISA table): VGPR0 = {K0 lanes0-15 | K2 lanes16-31}, VGPR1 = {K1 | K3}.
// f32 B layout (extrapolated from 8/16-bit B tables): lane half splits K the same way.
__global__ __launch_bounds__(256) void sa_conv_kernel(
    const float* __restrict__ avgp, const float* __restrict__ maxp,
    const float* __restrict__ w,    // [2][7][7]
    const float* __restrict__ bias, // [1]
    float* __restrict__ att) {
  __shared__ float rows[14][144];   // [plane*7 + ky][x tile: xb-3 .. xb+130]
  __shared__ float wl[112];         // [c][ky][kx(8)] zero-padded kernel

  const int tid = threadIdx.x;
  const int b = blockIdx.z, y = blockIdx.y, xb = blockIdx.x * 128;

  if (tid < 112) {
    int c  = tid / 56;
    int r  = (tid % 56) / 8;
    int kx = tid & 7;
    wl[tid] = (kx < 7) ? w[c * 49 + r * 7 + kx] : 0.0f;
  }
  for (int idx = tid; idx < 14 * 136; idx += 256) {
    int r     = idx / 136;          // 0..13
    int xi    = idx - r * 136;      // 0..135 (need 0..133)
    int plane = r / 7;
    int ky    = r - plane * 7;
    int gy    = y + ky - 3;
    int gx    = xb + xi - 3;
    float v = 0.0f;
    if ((unsigned)gy < (unsigned)Hn && (unsigned)gx < (unsigned)Wn)
      v = (plane ? maxp : avgp)[(size_t)b * HW + gy * Wn + gx];
    rows[r][xi] = v;
  }
  __syncthreads();

  const int wave = tid >> 5;
  const int lane = tid & 31;
  const int n    = lane & 15;
  const int half = lane >> 4;       // which K pair this lane half carries

  v8f acc = {0.f, 0.f, 0.f, 0.f, 0.f, 0.f, 0.f, 0.f};
  #pragma unroll
  for (int c = 0; c < 2; ++c) {
    #pragma unroll
    for (int ky = 0; ky < 7; ++ky) {
      const float* rp = &rows[c * 7 + ky][wave * 16 + n + half * 2];
      const float* wp = &wl[c * 56 + ky * 8 + half * 2];
      #pragma unroll
      for (int kxb = 0; kxb < 2; ++kxb) {
        v2f A = { wp[kxb * 4], wp[kxb * 4 + 1] };           // weights (bcast over M)
        v2f Bm = { rp[kxb * 4], rp[kxb * 4 + 1] };          // patch  (N = pixel)
        acc = __builtin_amdgcn_wmma_f32_16x16x4_f32(
            false, A, false, Bm, (short)0, acc, false, false);
      }
    }
  }
  if (lane < 16) {                  // every lane's acc[0] = result of pixel n
    float s = acc[0] + bias[0];
    float a = 1.0f / (1.0f + __expf(-s));
    att[(size_t)b * HW + (size_t)y * Wn + xb + wave * 16 + n] = a;
  }
}

// ---------------- Kernel C: out = x * att (bandwidth-bound) ------------------
__global__ __launch_bounds__(256) void sa_mul_kernel(
    const float* __restrict__ x, const float* __restrict__ att,
    float* __restrict__ out) {
  size_t idx = (size_t)blockIdx.x * 256 + threadIdx.x;  // 16,777,216 float4s
  int b    = (int)(idx >> 20);                          // / (CHW/4)
  int rem  = (int)(idx & ((CHW / 4) - 1));
  int pix4 = rem & ((HW / 4) - 1);
  float4 xv = ((const float4*)x)[idx];
  float4 av = ((const float4*)(att + (size_t)b * HW))[pix4];  // L2-resident reuse
  float4 o;
  o.x = xv.x * av.x; o.y = xv.y * av.y; o.z = xv.z * av.z; o.w = xv.w * av.w;
  ((float4*)out)[idx] = o;
}

extern "C" void kernel_launch(void* const* d_in, const int* in_sizes, int n_in,
                              void* d_out, int out_size, void* d_ws, size_t ws_size,
                              hipStream_t stream) {
  (void)in_sizes; (void)n_in; (void)out_size; (void)ws_size;
  const float* x    = (const float*)d_in[0];
  const float* w    = (const float*)d_in[1];   // 98 floats
  const float* bias = (const float*)d_in[2];   // 1 float
  float* out = (float*)d_out;

  float* avgp = (float*)d_ws;                  // 4 MB
  float* maxp = avgp + (size_t)Bn * HW;        // 4 MB
  float* att  = maxp + (size_t)Bn * HW;        // 4 MB  (12 MB scratch total)

  sa_reduce_kernel<<<1024, 256, 0, stream>>>(x, avgp, maxp);
  sa_conv_kernel<<<dim3(2, 256, 16), 256, 0, stream>>>(avgp, maxp, w, bias, att);
  sa_mul_kernel<<<65536, 256, 0, stream>>>(x, att, out);
}